// FactoredAttention_68126771249637
// MI455X (gfx1250) — compile-verified
//
#include <hip/hip_runtime.h>
#include <math.h>

// ---------------------------------------------------------------------------
// FactoredAttention on MI455X (gfx1250, wave32)
//   x(16384x1024) @ Wqkv(1024x3072) + b  -> qkv          [WMMA f32 16x16x4]
//   column causal attention over NB=32, 512 sequences     [VALU, ~1.5% FLOPs]
//   attn(16384x1024) @ Wout(1024x1024) + b -> out         [WMMA f32 16x16x4]
//
// GEMM: block tile 64x128, 4 waves, wave tile 32x64 (2x4 WMMA frags),
// BK=32, double-buffered LDS fed by GLOBAL_LOAD_ASYNC_TO_LDS_B128 when the
// toolchain exposes it (ASYNCcnt-tracked, overlaps with WMMA issue).
// ---------------------------------------------------------------------------

typedef float v2f __attribute__((ext_vector_type(2)));
typedef float v8f __attribute__((ext_vector_type(8)));
typedef int   v4i __attribute__((vector_size(16)));

#define BM 64
#define BN 128
#define BK 32
#define LDA_S 36    // LDS row stride (floats): 16B-aligned, conflict-free frag reads
#define LDB_S 136   // (2*136)%64 == 16 -> half-waves hit disjoint bank sets

#if defined(__gfx1250__) && __has_builtin(__builtin_amdgcn_global_load_async_to_lds_b128)
#define HAS_ASYNC_LDS 1
#else
#define HAS_ASYNC_LDS 0
#endif

#if HAS_ASYNC_LDS
typedef v4i __attribute__((address_space(1))) gv4i;  // global
typedef v4i __attribute__((address_space(3))) lv4i;  // LDS
#endif

__device__ __forceinline__ void cp16(const float* g, float* l) {
#if HAS_ASYNC_LDS
  __builtin_amdgcn_global_load_async_to_lds_b128(
      (gv4i*)(const_cast<float*>(g)), (lv4i*)l, 0, 0);
#else
  *(float4*)l = *(const float4*)g;
#endif
}

template <int N>
__device__ __forceinline__ void wait_async() {
#if HAS_ASYNC_LDS
#if __has_builtin(__builtin_amdgcn_s_wait_asynccnt)
  __builtin_amdgcn_s_wait_asynccnt((short)N);
#else
  asm volatile("s_wait_asynccnt %0" ::"n"(N) : "memory");
#endif
#endif
}

// Issue one chunk's staging (12 x B128 per thread: 4 for A tile, 8 for B tile).
__device__ __forceinline__ void stage_tiles(const float* __restrict__ A,
                                            const float* __restrict__ B,
                                            int K, int N, int m0, int n0, int k0,
                                            float* As, float* Bs, int tid) {
  // A tile: 64 rows x 32 cols; thread t -> row t>>1, 16-float half-row.
  const int arow = tid >> 1;
  const int acol = (tid & 1) * 16;
  const float* ag = A + (size_t)(m0 + arow) * K + k0 + acol;
  float* al = As + arow * LDA_S + acol;
#pragma unroll
  for (int i = 0; i < 4; ++i) cp16(ag + 4 * i, al + 4 * i);

  // B tile: 32 rows x 128 cols; thread t -> row t>>2, 32-float quarter-row.
  const int brow = tid >> 2;
  const int bcol = (tid & 3) * 32;
  const float* bg = B + (size_t)(k0 + brow) * N + n0 + bcol;
  float* bl = Bs + brow * LDB_S + bcol;
#pragma unroll
  for (int i = 0; i < 8; ++i) cp16(bg + 4 * i, bl + 4 * i);
}

// C(MxN) = A(MxK) @ B(KxN) + bias, row-major fp32.
__global__ __launch_bounds__(128)
void gemm_f32_wmma(const float* __restrict__ A, const float* __restrict__ B,
                   const float* __restrict__ bias, float* __restrict__ C,
                   int M, int N, int K) {
  __shared__ float As[2][BM * LDA_S];
  __shared__ float Bs[2][BK * LDB_S];

  const int tid   = threadIdx.x;
  const int lane  = tid & 31;
  const int wave  = tid >> 5;           // 0..3  (2 in M x 2 in N)
  const int waveM = (wave >> 1) * 32;
  const int waveN = (wave & 1) * 64;

  const int n0 = blockIdx.x * BN;
  const int m0 = blockIdx.y * BM;

  const int half = lane >> 4;
  const int l16  = lane & 15;

  v8f c[2][4];
#pragma unroll
  for (int fm = 0; fm < 2; ++fm)
#pragma unroll
    for (int fn = 0; fn < 4; ++fn)
      c[fm][fn] = (v8f){0.f, 0.f, 0.f, 0.f, 0.f, 0.f, 0.f, 0.f};

  // Prologue: stage chunk 0 into buffer 0.
  stage_tiles(A, B, K, N, m0, n0, 0, As[0], Bs[0], tid);

  for (int k0 = 0; k0 < K; k0 += BK) {
    const int cur = (k0 / BK) & 1;

    if (k0 + BK < K) {   // uniform branch: EXEC stays all-ones for WMMA
      stage_tiles(A, B, K, N, m0, n0, k0 + BK, As[cur ^ 1], Bs[cur ^ 1], tid);
      wait_async<12>();  // retire previous chunk, leave the 12 new in flight
    } else {
      wait_async<0>();
    }
    __syncthreads();     // all waves' staging of buf[cur] visible

    const float* Asb = As[cur];
    const float* Bsb = Bs[cur];
#pragma unroll
    for (int kk = 0; kk < BK; kk += 4) {
      // ISA layouts (wave32): for 16x4 f32 A and 4x16 B, VGPR v of a lane
      // holds k = kk + 2*half + v.  A: row = l16 (M).  B: col = l16 (N).
      const int kfrag = kk + 2 * half;

      v2f a[2], b[4];
#pragma unroll
      for (int fm = 0; fm < 2; ++fm) {
        const float* ap = Asb + (waveM + 16 * fm + l16) * LDA_S + kfrag;
        a[fm].x = ap[0];
        a[fm].y = ap[1];
      }
#pragma unroll
      for (int fn = 0; fn < 4; ++fn) {
        const float* bp = Bsb + kfrag * LDB_S + waveN + 16 * fn + l16;
        b[fn].x = bp[0];
        b[fn].y = bp[LDB_S];
      }

#pragma unroll
      for (int fm = 0; fm < 2; ++fm)
#pragma unroll
        for (int fn = 0; fn < 4; ++fn)
          c[fm][fn] = __builtin_amdgcn_wmma_f32_16x16x4_f32(
              false, a[fm], false, b[fn], (short)0, c[fm][fn], false, false);
    }
    __syncthreads();     // buf[cur] free before next iteration overwrites it
  }

  // Epilogue: bias add + store.  C/D layout: VGPR r -> row r + 8*half, col l16.
#pragma unroll
  for (int fn = 0; fn < 4; ++fn) {
    const int nc = n0 + waveN + 16 * fn + l16;
    const float bv = bias[nc];
#pragma unroll
    for (int fm = 0; fm < 2; ++fm) {
      const int rbase = m0 + waveM + 16 * fm + 8 * half;
#pragma unroll
      for (int r = 0; r < 8; ++r)
        C[(size_t)(rbase + r) * N + nc] = c[fm][fn][r] + bv;
    }
  }
}

// ---------------------------------------------------------------------------
// Column causal attention: sequences indexed by (b, bl) of length NB=32,
// head dim 1024.  One wave per (b, bl, q-position): 16384 waves.
// qkv row for (b, nb, bl) = ((b*32 + nb)*128 + bl), cols: q[0,1024) k[1024,2048)
// v[2048,3072).  Per-sequence working set is 384 KB -> resident in 192 MB L2.
// ---------------------------------------------------------------------------
__global__ __launch_bounds__(256)
void col_attn(const float* __restrict__ qkv, float* __restrict__ attn) {
  const int lane = threadIdx.x & 31;
  const int w    = blockIdx.x * 8 + (threadIdx.x >> 5);  // global wave id
  const int qn   = w & 31;          // query position along NB (causal axis)
  const int bl   = (w >> 5) & 127;  // within-block position
  const int b    = w >> 12;         // batch

  const size_t rowQ = (size_t)(b * 32 + qn) * 128 + bl;
  const float* Q = qkv + rowQ * 3072;
  const float scale = 0.03125f;     // 1/sqrt(1024)

  // Scores: lane kn ends up holding score[qn][kn]; masked lanes stay -inf.
  float sc = -INFINITY;
  for (int kn = 0; kn <= qn; ++kn) {
    const float* Kr = qkv + ((size_t)(b * 32 + kn) * 128 + bl) * 3072 + 1024;
    float acc = 0.f;
#pragma unroll 8
    for (int d = lane; d < 1024; d += 32)
      acc += Q[d] * Kr[d];                     // coalesced 128B per wave
#pragma unroll
    for (int off = 16; off; off >>= 1)
      acc += __shfl_xor(acc, off, 32);
    sc = (lane == kn) ? acc * scale : sc;      // lane-select, no divergence
  }

  // In-wave softmax over 32 lanes.
  float mx = sc;
#pragma unroll
  for (int off = 16; off; off >>= 1) mx = fmaxf(mx, __shfl_xor(mx, off, 32));
  float p = __expf(sc - mx);                   // -inf lanes -> 0
  float sum = p;
#pragma unroll
  for (int off = 16; off; off >>= 1) sum += __shfl_xor(sum, off, 32);
  p /= sum;

  // out[qn, :] = sum_k p[k] * V[k, :]; each lane owns 32 of the 1024 dims.
  float acc[32];
#pragma unroll
  for (int j = 0; j < 32; ++j) acc[j] = 0.f;

  for (int kn = 0; kn <= qn; ++kn) {
    const float pk = __shfl(p, kn, 32);
    const float* Vr = qkv + ((size_t)(b * 32 + kn) * 128 + bl) * 3072 + 2048;
#pragma unroll
    for (int j = 0; j < 32; ++j)
      acc[j] += pk * Vr[lane + 32 * j];
  }

  float* O = attn + rowQ * 1024;
#pragma unroll
  for (int j = 0; j < 32; ++j)
    O[lane + 32 * j] = acc[j];
}

// ---------------------------------------------------------------------------
extern "C" void kernel_launch(void* const* d_in, const int* in_sizes, int n_in,
                              void* d_out, int out_size, void* d_ws, size_t ws_size,
                              hipStream_t stream) {
  (void)in_sizes; (void)n_in; (void)out_size; (void)ws_size;

  const float* x    = (const float*)d_in[0];   // (4,4096,1024)
  const float* Wqkv = (const float*)d_in[1];   // (1024,3072)
  const float* bqkv = (const float*)d_in[2];   // (3072)
  const float* Wout = (const float*)d_in[3];   // (1024,1024)
  const float* bout = (const float*)d_in[4];   // (1024)
  float* out = (float*)d_out;                  // (4,4096,1024)

  float* qkv  = (float*)d_ws;                       // 16384 x 3072 f32 (192 MB)
  float* attn = qkv + (size_t)16384 * 3072;         // 16384 x 1024 f32 (64 MB)

  const int M = 16384, D = 1024;

  // Stage 1: QKV projection.
  gemm_f32_wmma<<<dim3((3 * D) / BN, M / BM), dim3(128), 0, stream>>>(
      x, Wqkv, bqkv, qkv, M, 3 * D, D);

  // Stage 2: column causal attention (16384 waves, 8 waves/block).
  col_attn<<<dim3(16384 / 8), dim3(256), 0, stream>>>(qkv, attn);

  // Stage 3: output projection.
  gemm_f32_wmma<<<dim3(D / BN, M / BM), dim3(128), 0, stream>>>(
      attn, Wout, bout, out, M, D, D);
}